// attention_46840913330459
// MI455X (gfx1250) — compile-verified
//
#include <hip/hip_runtime.h>
#include <hip/hip_bf16.h>

#define NSEG 8192
#define D    128

typedef __attribute__((ext_vector_type(2))) float v2f;
typedef __attribute__((ext_vector_type(4))) float v4f;
typedef __attribute__((ext_vector_type(8))) float v8f;

// ---- order-preserving float<->uint transform (for atomicMax on floats) ----
__device__ __forceinline__ unsigned orderedF32(float f) {
    unsigned u = __float_as_uint(f);
    return (u & 0x80000000u) ? ~u : (u | 0x80000000u);
}
__device__ __forceinline__ float unorderF32(unsigned u) {
    return (u & 0x80000000u) ? __uint_as_float(u & 0x7FFFFFFFu)
                             : __uint_as_float(~u);
}

// ---------------------------------------------------------------------------
// Kernel 0: zero out-pool region of d_out, segmax (encoded -inf == 0), denom
// ---------------------------------------------------------------------------
__global__ void init_kernel(float* __restrict__ out, unsigned* __restrict__ segmax,
                            float* __restrict__ denom, int outN) {
    int stride = gridDim.x * blockDim.x;
    for (int i = blockIdx.x * blockDim.x + threadIdx.x; i < outN; i += stride) {
        out[i] = 0.0f;
        if (i < NSEG) { segmax[i] = 0u; denom[i] = 0.0f; }
    }
}

// ---------------------------------------------------------------------------
// Kernel 1: gate = x @ W^T + b via V_WMMA_F32_16X16X4_F32, + segment max.
// Block = 128 threads (4 waves). Block handles 64 rows; each wave 16 rows.
// LDS tile 64x128 (row stride 132 floats -> conflict-free column reads).
// B operand is W in column N==0 only: selected by multiplying with a hoisted
// per-lane mask (avoids divergent EXEC-masked LDS loads in the WMMA loop).
// ---------------------------------------------------------------------------
__global__ __launch_bounds__(128)
void gate_wmma_kernel(const float* __restrict__ x, const int* __restrict__ batch,
                      const float* __restrict__ W, const float* __restrict__ bias,
                      float* __restrict__ gate, unsigned* __restrict__ segmax) {
    __shared__ float tile[64 * 132];
    __shared__ float w_s[D];

    const int t = threadIdx.x;
    const long long rowBase = (long long)blockIdx.x * 64;
    const v4f* xv = (const v4f*)(x + rowBase * D);

    // coalesced, non-temporal 16B tile load: 2048 vec4 = 64x128 floats
    #pragma unroll
    for (int j = 0; j < 16; ++j) {
        int e4  = j * 128 + t;           // vec4 index in tile
        int row = e4 >> 5;               // 32 vec4 per row
        int col = (e4 & 31) << 2;
        v4f v = __builtin_nontemporal_load(&xv[e4]);
        *(v4f*)&tile[row * 132 + col] = v;
    }
    w_s[t] = W[t];
    __syncthreads();

    const int wave = t >> 5;
    const int lane = t & 31;
    const int m    = lane & 15;          // row within 16-row wave tile
    const int half = lane >> 4;          // 0: K=0,1   1: K=2,3
    const float msk = (m == 0) ? 1.0f : 0.0f;  // column-0 selector for B
    const float* arow = &tile[(wave * 16 + m) * 132];
    const float* wrow = &w_s[half * 2];

    v8f c = {};
    #pragma unroll
    for (int ck = 0; ck < 32; ++ck) {    // 32 chunks of K=4 -> D=128
        const int k0 = ck * 4;
        v2f a, b;
        a.x = arow[k0 + half * 2];
        a.y = arow[k0 + half * 2 + 1];
        b.x = wrow[k0]     * msk;        // unconditional ds_load_b64 + mul
        b.y = wrow[k0 + 1] * msk;
        c = __builtin_amdgcn_wmma_f32_16x16x4_f32(
                /*neg_a=*/false, a, /*neg_b=*/false, b,
                /*c_mod=*/(short)0, c, /*reuse_a=*/false, /*reuse_b=*/false);
    }

    // Column N=0 lives in lanes 0 (rows 0..7) and 16 (rows 8..15): c[0..7]
    if (m == 0) {
        const float bb = bias[0];
        #pragma unroll
        for (int j = 0; j < 8; ++j) {
            long long grow = rowBase + wave * 16 + half * 8 + j;
            float g = c[j] + bb;
            gate[grow] = g;
            atomicMax(&segmax[batch[grow]], orderedF32(g));
        }
    }
}

// scalar tail for rows not covered by 64-row WMMA tiles (N % 64)
__global__ void gate_tail_kernel(const float* __restrict__ x, const int* __restrict__ batch,
                                 const float* __restrict__ W, const float* __restrict__ bias,
                                 float* __restrict__ gate, unsigned* __restrict__ segmax,
                                 int start, int N) {
    int i = start + blockIdx.x * blockDim.x + threadIdx.x;
    if (i >= N) return;
    float acc = bias[0];
    const float* xr = x + (long long)i * D;
    for (int k = 0; k < D; ++k) acc += xr[k] * W[k];
    gate[i] = acc;
    atomicMax(&segmax[batch[i]], orderedF32(acc));
}

// ---------------------------------------------------------------------------
// Kernel 2: e = exp(gate - gmax[batch]); denom[seg] += e
// ---------------------------------------------------------------------------
__global__ void softmax_e_kernel(const float* __restrict__ gate, const int* __restrict__ batch,
                                 const unsigned* __restrict__ segmax,
                                 float* __restrict__ e, float* __restrict__ denom, int N) {
    int i = blockIdx.x * blockDim.x + threadIdx.x;
    if (i >= N) return;
    int s = batch[i];
    float ev = __expf(gate[i] - unorderF32(segmax[s]));
    e[i] = ev;
    atomicAdd(&denom[s], ev);
}

// ---------------------------------------------------------------------------
// Kernel 3: attn = e/(denom+eps); out[s,:] += attn * relu(x).
// 128 threads = one feature column each; register accumulator per column,
// flushed with atomicAdd only at sorted-segment boundaries.
// ---------------------------------------------------------------------------
__global__ __launch_bounds__(128)
void pool_kernel(const float* __restrict__ x, const int* __restrict__ batch,
                 const float* __restrict__ e, const float* __restrict__ denom,
                 float* __restrict__ out, float* __restrict__ attn,
                 int N, int rowsPerBlock) {
    const int t = threadIdx.x;
    long long r0 = (long long)blockIdx.x * rowsPerBlock;
    long long r1 = r0 + rowsPerBlock;
    if (r1 > N) r1 = N;
    if (r0 >= r1) return;

    int   prev = batch[r0];
    float acc  = 0.0f;
    for (long long i = r0; i < r1; ++i) {
        int s = batch[i];
        if (s != prev) {
            atomicAdd(&out[(long long)prev * D + t], acc);
            acc = 0.0f;
            prev = s;
        }
        float ai = e[i] / (denom[s] + 1e-16f);
        if (t == 0) attn[i] = ai;
        float v = __builtin_nontemporal_load(&x[i * (long long)D + t]);
        acc = fmaf(ai, fmaxf(v, 0.0f), acc);
    }
    atomicAdd(&out[(long long)prev * D + t], acc);
}

// ---------------------------------------------------------------------------
extern "C" void kernel_launch(void* const* d_in, const int* in_sizes, int n_in,
                              void* d_out, int out_size, void* d_ws, size_t ws_size,
                              hipStream_t stream) {
    const float* x     = (const float*)d_in[0];   // [N, 128]
    const int*   batch = (const int*)d_in[1];     // [N] sorted segment ids
    const float* W     = (const float*)d_in[2];   // [1, 128]
    const float* bias  = (const float*)d_in[3];   // [1]
    const int N = in_sizes[1];

    float* out  = (float*)d_out;                  // [NSEG*128]
    float* attn = out + (size_t)NSEG * D;         // [N]

    // scratch layout in d_ws
    float*    gate   = (float*)d_ws;              // [N]
    float*    e      = gate + N;                  // [N]
    unsigned* segmax = (unsigned*)(e + N);        // [NSEG]
    float*    denom  = (float*)(segmax + NSEG);   // [NSEG]

    // 0) init pooled output + segment stats
    init_kernel<<<1024, 256, 0, stream>>>(out, segmax, denom, NSEG * D);

    // 1) WMMA gate GEMV + segment max
    int nb1 = N / 64;
    if (nb1 > 0)
        gate_wmma_kernel<<<nb1, 128, 0, stream>>>(x, batch, W, bias, gate, segmax);
    int rem = N % 64;
    if (rem > 0)
        gate_tail_kernel<<<1, 64, 0, stream>>>(x, batch, W, bias, gate, segmax,
                                               N - rem, N);

    // 2) stable exp + segment sum
    softmax_e_kernel<<<(N + 255) / 256, 256, 0, stream>>>(gate, batch, segmax,
                                                          e, denom, N);

    // 3) fused attention + ReLU-weighted pooling
    const int rows3 = 1024;
    int nb3 = (N + rows3 - 1) / rows3;
    pool_kernel<<<nb3, 128, 0, stream>>>(x, batch, e, denom, out, attn, N, rows3);
}